// GraphAttentionLayer_48713519071262
// MI455X (gfx1250) — compile-verified
//
#include <hip/hip_runtime.h>
#include <math.h>

// GAT layer, single-pass flash-softmax over 200000x128 f32 rows.
// WMMA f32 16x16x4 computes 16 row-logits per wave per K-chain (exact f32).
// h_j streamed from HBM exactly once (102.4 MB -> ~4.4us roofline @23.3TB/s),
// staged to LDS via async-to-LDS loads with double buffering when available.

typedef __attribute__((ext_vector_type(2))) float v2f;
typedef __attribute__((ext_vector_type(8))) float v8f;
typedef int v4i __attribute__((vector_size(4 * sizeof(int))));

#ifndef __has_builtin
#define __has_builtin(x) 0
#endif
#if __has_builtin(__builtin_amdgcn_global_load_async_to_lds_b128) && \
    __has_builtin(__builtin_amdgcn_s_wait_asynccnt)
#define USE_ASYNC_LDS 1
#else
#define USE_ASYNC_LDS 0
#endif

#define IN_F      128
#define TILE_ROWS 128
#define PITCH     132   // 128 + 4 pad: kills LDS bank conflicts for row-gathers
#define NWAVES    8
#define PART_SZ   130   // per-partial: m, s, acc[128]

// Stage one 128x128 f32 tile into LDS. Rows beyond nrows are clamped to the
// last valid row (never zero-filled): this keeps EXEC all-ones so the async
// path issues exactly 16 ASYNCcnt increments per wave per tile, and clamped
// garbage rows are provably ignored (each WMMA C-row depends only on its own
// A-row, and the softmax loop only touches j < valid).
__device__ __forceinline__ void stage_tile(float* __restrict__ buf,
                                           const float* __restrict__ hj,
                                           int row0, int nrows, int t)
{
#pragma unroll
    for (int i = 0; i < 16; ++i) {
        int idx = i * 256 + t;          // float4 index within tile
        int r   = idx >> 5;             // row within tile
        int c4  = (idx & 31) << 2;      // column (multiple of 4)
        int gr  = row0 + r;
        if (gr > nrows - 1) gr = nrows - 1;
        const float* gp = &hj[(size_t)gr * IN_F + c4];
        float*       lp = &buf[r * PITCH + c4];
#if USE_ASYNC_LDS
        __builtin_amdgcn_global_load_async_to_lds_b128(
            (__attribute__((address_space(1))) v4i*)gp,
            (__attribute__((address_space(3))) v4i*)lp,
            0, 0);
#else
        *(float4*)lp = *(const float4*)gp;
#endif
    }
}

__global__ __launch_bounds__(256) void gat_pass1(
    const float* __restrict__ hj, const float* __restrict__ a,
    float* __restrict__ ws, int nrows, int ntiles)
{
    __shared__ float tilebuf[2][TILE_ROWS * PITCH];   // 2 x 67584 B
    __shared__ float comb[NWAVES * PART_SZ];          //     4160 B
    __shared__ float aj[IN_F];                        //      512 B

    const int t     = threadIdx.x;
    const int lane  = t & 31;
    const int wave  = t >> 5;
    const int lhalf = lane >> 4;   // 0 for lanes 0-15, 1 for lanes 16-31
    const int mrow  = lane & 15;   // A-matrix row index M for this lane

    if (t < IN_F) aj[t] = a[IN_F + t];

    // Prologue of the software pipeline: issue tile 0 before touching aj reads
    int ti = blockIdx.x;
    if (ti < ntiles) stage_tile(tilebuf[0], hj, ti * TILE_ROWS, nrows, t);

    __syncthreads();   // aj visible (async tile waits handled in the loop)

    // B fragments for all 32 K-steps: B[k][n] = aj[4*kstep + k], broadcast over N.
    // Assumed f32 4x16 B layout (mirror of documented A 16x4):
    //   VGPR r, lane l  ->  K = r + 2*(l>=16), N = l%16
    v2f bf[32];
#pragma unroll
    for (int k = 0; k < 32; ++k) {
        const float* p = &aj[4 * k + 2 * lhalf];
        bf[k].x = p[0];
        bf[k].y = p[1];
    }

    float m = -INFINITY, s = 0.f;
    float acc0 = 0.f, acc1 = 0.f, acc2 = 0.f, acc3 = 0.f;

    int cur = 0;
    for (; ti < ntiles; ti += gridDim.x, cur ^= 1) {
        // ---- issue next tile's async loads, then wait for current tile ----
        const int tnext = ti + gridDim.x;
        if (tnext < ntiles) {
            stage_tile(tilebuf[cur ^ 1], hj, tnext * TILE_ROWS, nrows, t);
#if USE_ASYNC_LDS
            __builtin_amdgcn_s_wait_asynccnt(16);   // current 16 done (in-order)
#endif
        } else {
#if USE_ASYNC_LDS
            __builtin_amdgcn_s_wait_asynccnt(0);
#endif
        }
        __syncthreads();   // every wave waited on its own asyncs -> tile ready

        const float* tb = tilebuf[cur];
        const int wrow   = wave * 16;
        const int gfirst = ti * TILE_ROWS + wrow;
        int rem   = nrows - gfirst;
        int valid = rem > 16 ? 16 : rem;
        if (valid > 0) {   // wave-uniform branch: EXEC all-ones inside (WMMA req)
            v8f c = {0.f, 0.f, 0.f, 0.f, 0.f, 0.f, 0.f, 0.f};
            // ---- logits for 16 rows: chain of 32 x V_WMMA_F32_16X16X4_F32 ----
#pragma unroll
            for (int k = 0; k < 32; ++k) {
                const float* p = &tb[(wrow + mrow) * PITCH + 4 * k + 2 * lhalf];
                v2f af;
                af.x = p[0];
                af.y = p[1];
                c = __builtin_amdgcn_wmma_f32_16x16x4_f32(
                        false, af, false, bf[k], (short)0, c, false, false);
            }
            // All N columns of C are equal -> row j logit lives in c[j%8] of
            // lane 0 (rows 0-7) / lane 16 (rows 8-15).
            float lg[16];
#pragma unroll
            for (int j = 0; j < 8; ++j) {
                lg[j]     = __int_as_float(__builtin_amdgcn_readlane(__float_as_int(c[j]), 0));
                lg[8 + j] = __int_as_float(__builtin_amdgcn_readlane(__float_as_int(c[j]), 16));
            }
            // ---- online softmax update ----
            float tmax = -INFINITY;
            for (int j = 0; j < valid; ++j) tmax = fmaxf(tmax, lg[j]);
            if (tmax > m) {
                float sc = __expf(m - tmax);
                s *= sc; acc0 *= sc; acc1 *= sc; acc2 *= sc; acc3 *= sc;
                m = tmax;
            }
            for (int j = 0; j < valid; ++j) {
                float w = __expf(lg[j] - m);
                s += w;
                const float* rp = &tb[(wrow + j) * PITCH];   // conflict-free
                acc0 += w * rp[lane];
                acc1 += w * rp[lane + 32];
                acc2 += w * rp[lane + 64];
                acc3 += w * rp[lane + 96];
            }
        }
        __syncthreads();   // all waves done reading tilebuf[cur] before reuse
    }

    // ---- per-wave partials -> LDS ----
    if (lane == 0) { comb[wave * PART_SZ] = m; comb[wave * PART_SZ + 1] = s; }
    comb[wave * PART_SZ + 2 + lane]      = acc0;
    comb[wave * PART_SZ + 2 + lane + 32] = acc1;
    comb[wave * PART_SZ + 2 + lane + 64] = acc2;
    comb[wave * PART_SZ + 2 + lane + 96] = acc3;
    __syncthreads();

    // ---- block combine: 8 wave partials -> 1 block partial in ws ----
    if (t < IN_F) {
        float M = -INFINITY;
#pragma unroll
        for (int w = 0; w < NWAVES; ++w) M = fmaxf(M, comb[w * PART_SZ]);
        float S = 0.f, V = 0.f;
#pragma unroll
        for (int w = 0; w < NWAVES; ++w) {
            float mw = comb[w * PART_SZ];
            float sc = (mw < -1e37f) ? 0.f : __expf(mw - M);
            S += sc * comb[w * PART_SZ + 1];
            V += sc * comb[w * PART_SZ + 2 + t];
        }
        if (t == 0) {
            ws[(size_t)blockIdx.x * PART_SZ]     = M;
            ws[(size_t)blockIdx.x * PART_SZ + 1] = S;
        }
        ws[(size_t)blockIdx.x * PART_SZ + 2 + t] = V;
    }
}

__global__ __launch_bounds__(128) void gat_pass2(
    const float* __restrict__ ws, float* __restrict__ out, int nb)
{
    const int f = threadIdx.x;
    float M = -INFINITY;
    for (int b = 0; b < nb; ++b) M = fmaxf(M, ws[(size_t)b * PART_SZ]);
    float S = 0.f, V = 0.f;
    for (int b = 0; b < nb; ++b) {
        float mb = ws[(size_t)b * PART_SZ];
        float sc = (mb < -1e37f) ? 0.f : __expf(mb - M);
        S += sc * ws[(size_t)b * PART_SZ + 1];
        V += sc * ws[(size_t)b * PART_SZ + 2 + f];
    }
    float h = V / S;
    out[f] = (h > 0.f) ? h : expm1f(h);   // ELU, alpha=1
}

extern "C" void kernel_launch(void* const* d_in, const int* in_sizes, int n_in,
                              void* d_out, int out_size, void* d_ws, size_t ws_size,
                              hipStream_t stream)
{
    // setup_inputs order: h_i (unused: its logit term is constant and cancels
    // exactly in softmax), h_j, a
    const float* hj = (const float*)d_in[1];
    const float* a  = (const float*)d_in[2];
    float* out = (float*)d_out;
    float* ws  = (float*)d_ws;

    int nrows  = in_sizes[1] / IN_F;
    int ntiles = (nrows + TILE_ROWS - 1) / TILE_ROWS;

    int nblk = 512;
    if (nblk > ntiles) nblk = ntiles;
    int maxblk = (int)(ws_size / (PART_SZ * sizeof(float)));
    if (nblk > maxblk) nblk = maxblk;
    if (nblk < 1) nblk = 1;

    gat_pass1<<<nblk, 256, 0, stream>>>(hj, a, ws, nrows, ntiles);
    gat_pass2<<<1, 128, 0, stream>>>(ws, out, nblk);
}